// MultiBoxLoss_59579786330818
// MI455X (gfx1250) — compile-verified
//
#include <hip/hip_runtime.h>
#include <math.h>

#define Bn 64
#define Pn 24564
#define Cn 81
#define On 20
#define THRESH 0.5f
#define NEGR 3

typedef __attribute__((ext_vector_type(2))) float v2f;
typedef __attribute__((ext_vector_type(8))) float v8f;

__device__ __forceinline__ float sl1(float x) {
    float a = fabsf(x);
    return (a < 1.0f) ? 0.5f * a * a : a - 0.5f;
}

// ---------------------------------------------------------------------------
// Kernel 0: zero the small accumulators
// ---------------------------------------------------------------------------
__global__ void init_kernel(float* __restrict__ accf, int* __restrict__ total_pos,
                            int* __restrict__ n_pos) {
    int t = threadIdx.x;
    if (t < 4) accf[t] = 0.0f;
    if (t == 0) *total_pos = 0;
    if (t < Bn) n_pos[t] = 0;
}

// ---------------------------------------------------------------------------
// Kernel 1: per-batch prior<->box matching (IoU, bidirectional argmax, override)
// One block per batch row. Writes matched[b,p] = obj | (label<<8), n_pos[b].
// Uses ovbuf (later overwritten by conf_all) as overlap scratch.
// ---------------------------------------------------------------------------
__global__ __launch_bounds__(256) void match_kernel(
    const float* __restrict__ boxes, const int* __restrict__ labels,
    const float* __restrict__ priors, int* __restrict__ matched,
    float* __restrict__ ovbuf, int* __restrict__ n_pos, int* __restrict__ total_pos)
{
    int b = blockIdx.x;
    int tid = threadIdx.x;

    __shared__ float bx0[On], by0[On], bx1[On], by1[On], barea[On];
    __shared__ int   blab[On];
    __shared__ float rv[On * 256];
    __shared__ int   ri[On * 256];
    __shared__ int   prior_fo[On];
    __shared__ int   s_cnt;

    if (tid < On) {
        const float* bb = boxes + ((size_t)b * On + tid) * 4;
        float x0 = bb[0], y0 = bb[1], x1 = bb[2], y1 = bb[3];
        bx0[tid] = x0; by0[tid] = y0; bx1[tid] = x1; by1[tid] = y1;
        barea[tid] = (x1 - x0) * (y1 - y0);
        blab[tid] = labels[(size_t)b * On + tid];
    }
    if (tid == 0) s_cnt = 0;
    __syncthreads();

    float obest[On];
    int   oidx[On];
#pragma unroll
    for (int o = 0; o < On; ++o) { obest[o] = -1.0f; oidx[o] = 0; }

    for (int p = tid; p < Pn; p += 256) {
        const float* pr = priors + (size_t)p * 4;
        float cx = pr[0], cy = pr[1], w = pr[2], h = pr[3];
        float px0 = cx - 0.5f * w, py0 = cy - 0.5f * h;
        float px1 = cx + 0.5f * w, py1 = cy + 0.5f * h;
        float parea = w * h;
        float bestv = -1.0f; int besto = 0;
#pragma unroll
        for (int o = 0; o < On; ++o) {
            float lx = fmaxf(bx0[o], px0), ly = fmaxf(by0[o], py0);
            float rx = fminf(bx1[o], px1), ry = fminf(by1[o], py1);
            float iw = fmaxf(rx - lx, 0.0f), ih = fmaxf(ry - ly, 0.0f);
            float inter = iw * ih;
            float iou = inter / (barea[o] + parea - inter);
            if (iou > bestv) { bestv = iou; besto = o; }     // first-max over o
            if (iou > obest[o]) { obest[o] = iou; oidx[o] = p; }
        }
        ovbuf[(size_t)b * Pn + p] = bestv;
        matched[(size_t)b * Pn + p] = besto;
    }

#pragma unroll
    for (int o = 0; o < On; ++o) { rv[o * 256 + tid] = obest[o]; ri[o * 256 + tid] = oidx[o]; }
    __syncthreads();

    if (tid < On) {   // per-object argmax over all priors, first-index tie-break
        float bv = rv[tid * 256]; int bi = ri[tid * 256];
        for (int t = 1; t < 256; ++t) {
            float v = rv[tid * 256 + t]; int i = ri[tid * 256 + t];
            if (v > bv || (v == bv && i < bi)) { bv = v; bi = i; }
        }
        prior_fo[tid] = bi;
    }
    __syncthreads();

    // pass 2: apply override (last-wins over o), threshold labels, count positives
    int cnt = 0;
    for (int p = tid; p < Pn; p += 256) {
        float ov = ovbuf[(size_t)b * Pn + p];
        int obj = matched[(size_t)b * Pn + p];
#pragma unroll
        for (int o = 0; o < On; ++o)
            if (prior_fo[o] == p) { obj = o; ov = 1.0f; }
        int lab = (ov < THRESH) ? 0 : blab[obj];
        cnt += (lab != 0);
        matched[(size_t)b * Pn + p] = obj | (lab << 8);
    }
    atomicAdd(&s_cnt, cnt);
    __syncthreads();
    if (tid == 0) { n_pos[b] = s_cnt; atomicAdd(total_pos, s_cnt); }
}

// ---------------------------------------------------------------------------
// Kernel 2: one wave32 per prior. Cross-entropy over 81 classes (coalesced:
// lane reads class lane, lane+32, lane+64) + smooth-L1 loc loss on positives.
// ---------------------------------------------------------------------------
__global__ __launch_bounds__(256) void conf_kernel(
    const float* __restrict__ locs, const float* __restrict__ scores,
    const float* __restrict__ boxes, const float* __restrict__ priors,
    const int* __restrict__ matched,
    float* __restrict__ conf_all, float* __restrict__ conf_neg,
    float* __restrict__ loc_arr)
{
    int w = blockIdx.x * 8 + (threadIdx.x >> 5);
    int lane = threadIdx.x & 31;
    if (w >= Bn * Pn) return;

    size_t base = (size_t)w * Cn;
    __builtin_prefetch((const void*)(scores + base + 8 * Cn), 0, 0);

    float v0 = scores[base + lane];
    float v1 = scores[base + 32 + lane];
    float v2 = (lane < 17) ? scores[base + 64 + lane] : -__builtin_inff();

    float m = fmaxf(fmaxf(v0, v1), v2);
    for (int off = 16; off > 0; off >>= 1) m = fmaxf(m, __shfl_xor(m, off, 32));

    float e = expf(v0 - m) + expf(v1 - m) + ((lane < 17) ? expf(v2 - m) : 0.0f);
    for (int off = 16; off > 0; off >>= 1) e += __shfl_xor(e, off, 32);
    float lse = logf(e) + m;

    if (lane == 0) {
        int mt = matched[w];
        int obj = mt & 0xFF;
        int lab = mt >> 8;
        float conf = lse - scores[base + lab];
        conf_all[w] = conf;
        conf_neg[w] = (lab != 0) ? 0.0f : conf;

        float le = 0.0f;
        if (lab != 0) {
            int b = w / Pn;
            int p = w - b * Pn;
            const float* bb = boxes + ((size_t)b * On + obj) * 4;
            float bcx = 0.5f * (bb[0] + bb[2]);
            float bcy = 0.5f * (bb[1] + bb[3]);
            float bw = bb[2] - bb[0];
            float bh = bb[3] - bb[1];
            const float* pr = priors + (size_t)p * 4;
            float g0 = (bcx - pr[0]) / (pr[2] * 0.1f);
            float g1 = (bcy - pr[1]) / (pr[3] * 0.1f);
            float g2 = logf(bw / pr[2]) * 5.0f;
            float g3 = logf(bh / pr[3]) * 5.0f;
            const float* pl = locs + (size_t)w * 4;
            le = sl1(pl[0] - g0) + sl1(pl[1] - g1) + sl1(pl[2] - g2) + sl1(pl[3] - g3);
        }
        loc_arr[w] = le;
    }
}

// ---------------------------------------------------------------------------
// Kernel 3: tensor-core array reduction via V_WMMA_F32_16X16X4_F32.
// D = A(16x4) * ones(4x16) + C -> 16 row sums accumulated in f32 in C.
// Each wave sums tiles of 64 contiguous floats; grid.y selects the array.
// ---------------------------------------------------------------------------
__global__ __launch_bounds__(256) void wmma_reduce_kernel(
    const float* __restrict__ conf_all, const float* __restrict__ conf_neg,
    const float* __restrict__ loc_arr, float* __restrict__ accf)
{
    const float* src = (blockIdx.y == 0) ? conf_all
                     : (blockIdx.y == 1) ? conf_neg : loc_arr;

    int lane  = threadIdx.x & 31;
    int wave  = blockIdx.x * 8 + (threadIdx.x >> 5);
    int nwaves = gridDim.x * 8;
    const int ntiles = (Bn * Pn) / 64;   // 1572096 / 64 = 24564 exactly

    // A-matrix 16x4 f32 layout: lanes 0-15 hold K=0,1 ; lanes 16-31 hold K=2,3
    int m    = lane & 15;
    int koff = (lane >> 4) * 2;

    v2f bones = {1.0f, 1.0f};   // ones matrix (layout-invariant)
    v8f c = {};
    for (int t = wave; t < ntiles; t += nwaves) {
        const float* x = src + (size_t)t * 64 + m * 4 + koff;
        v2f a; a[0] = x[0]; a[1] = x[1];
        c = __builtin_amdgcn_wmma_f32_16x16x4_f32(
                /*neg_a=*/false, a, /*neg_b=*/false, bones,
                /*c_mod=*/(short)0, c, /*reuse_a=*/false, /*reuse_b=*/false);
    }
    // every row sum is replicated across 16 columns -> divide lane total by 16
    float partial = c[0] + c[1] + c[2] + c[3] + c[4] + c[5] + c[6] + c[7];
    for (int off = 16; off > 0; off >>= 1) partial += __shfl_xor(partial, off, 32);
    if (lane == 0) atomicAdd(&accf[blockIdx.y], partial * (1.0f / 16.0f));
}

// ---------------------------------------------------------------------------
// Kernel 4: per-batch descending bitonic sort of 24564 values padded to 32768
// in 128 KB LDS (CDNA5: 320 KB/WG), then sum of top min(3*n_pos, P).
// ---------------------------------------------------------------------------
#define SORTN 32768
__global__ __launch_bounds__(1024) void sort_topk_kernel(
    const float* __restrict__ conf_neg, const int* __restrict__ n_pos,
    float* __restrict__ accf)
{
    __shared__ float s[SORTN];          // 131072 bytes
    __shared__ float wsum[32];
    int b = blockIdx.x;
    int tid = threadIdx.x;

    for (int i = tid; i < SORTN; i += 1024)
        s[i] = (i < Pn) ? conf_neg[(size_t)b * Pn + i] : -1.0f;   // pad below all losses (>=0)
    __syncthreads();

    for (int k = 2; k <= SORTN; k <<= 1) {
        for (int j = k >> 1; j > 0; j >>= 1) {
            for (int t = tid; t < SORTN; t += 1024) {
                int l = t ^ j;
                if (l > t) {
                    bool up = ((t & k) == 0);   // descending segment
                    float x = s[t], y = s[l];
                    if ((x < y) == up) { s[t] = y; s[l] = x; }
                }
            }
            __syncthreads();
        }
    }

    int kk = NEGR * n_pos[b];
    if (kk > Pn) kk = Pn;
    float part = 0.0f;
    for (int i = tid; i < kk; i += 1024) part += s[i];
    for (int off = 16; off > 0; off >>= 1) part += __shfl_xor(part, off, 32);
    if ((tid & 31) == 0) wsum[tid >> 5] = part;
    __syncthreads();
    if (tid < 32) {
        float v = wsum[tid];
        for (int off = 16; off > 0; off >>= 1) v += __shfl_xor(v, off, 32);
        if (tid == 0) atomicAdd(&accf[3], v);
    }
}

// ---------------------------------------------------------------------------
// Kernel 5: scalar combine
// accf: [0]=sum(conf_all) [1]=sum(conf_neg) [2]=sum(loc) [3]=hard_neg_sum
// ---------------------------------------------------------------------------
__global__ void final_kernel(const float* __restrict__ accf,
                             const int* __restrict__ total_pos,
                             float* __restrict__ out)
{
    if (threadIdx.x == 0 && blockIdx.x == 0) {
        float conf_all_sum = accf[0];
        float conf_neg_sum = accf[1];
        float loc_sum      = accf[2];
        float hard_sum     = accf[3];
        int tp = *total_pos;
        float conf_pos_sum = conf_all_sum - conf_neg_sum;
        float conf_matched = (hard_sum + conf_pos_sum) / fmaxf((float)tp, 1.0f);
        float loc_loss     = loc_sum / fmaxf(4.0f * (float)tp, 1.0f);
        float loss = (tp > 0) ? (conf_matched + loc_loss)
                              : (conf_all_sum / (float)(Bn * Pn));
        out[0] = loss;
    }
}

// ---------------------------------------------------------------------------
extern "C" void kernel_launch(void* const* d_in, const int* in_sizes, int n_in,
                              void* d_out, int out_size, void* d_ws, size_t ws_size,
                              hipStream_t stream) {
    const float* locs   = (const float*)d_in[0];   // [B,P,4]
    const float* scores = (const float*)d_in[1];   // [B,P,C]
    const float* boxes  = (const float*)d_in[2];   // [B,O,4]
    const int*   labels = (const int*)d_in[3];     // [B,O]
    const float* priors = (const float*)d_in[4];   // [P,4]
    float* out = (float*)d_out;

    const size_t np = (size_t)Bn * Pn;
    char* base = (char*)d_ws;
    int*   matched  = (int*)  (base);               // np ints
    float* conf_all = (float*)(base + np * 4);      // np floats (also IoU scratch)
    float* conf_neg = (float*)(base + np * 8);      // np floats
    float* loc_arr  = (float*)(base + np * 12);     // np floats
    int*   n_pos    = (int*)  (base + np * 16);     // Bn ints
    float* accf     = (float*)(base + np * 16 + 256);
    int*   total_pos= (int*)  (base + np * 16 + 256 + 64);

    init_kernel<<<1, 64, 0, stream>>>(accf, total_pos, n_pos);

    match_kernel<<<Bn, 256, 0, stream>>>(boxes, labels, priors, matched,
                                         conf_all, n_pos, total_pos);

    int nblocks = (Bn * Pn + 7) / 8;   // one wave32 per prior, 8 waves/block
    conf_kernel<<<nblocks, 256, 0, stream>>>(locs, scores, boxes, priors, matched,
                                             conf_all, conf_neg, loc_arr);

    dim3 rg(256, 3);
    wmma_reduce_kernel<<<rg, 256, 0, stream>>>(conf_all, conf_neg, loc_arr, accf);

    sort_topk_kernel<<<Bn, 1024, 0, stream>>>(conf_neg, n_pos, accf);

    final_kernel<<<1, 1, 0, stream>>>(accf, total_pos, out);
}